// BottleneckBlock_79096117723783
// MI455X (gfx1250) — compile-verified
//
#include <hip/hip_runtime.h>
#include <math.h>

typedef __attribute__((ext_vector_type(2))) float v2f;
typedef __attribute__((ext_vector_type(8))) float v8f;
typedef int v4i __attribute__((vector_size(16)));   // matches builtin param type

#define NB      8
#define WDIM    64
#define TDIM    4096
#define KBINS   2048
#define NROWS   (NB * TDIM)          // 32768 rows of xf
#define ROWS_PER_BLOCK 128           // 8 waves * 16 rows
#define NTILES  (KBINS / 16)         // 128 codebook tiles
#define XD_OFF  NROWS
#define SCAL_OFF (NROWS + NROWS * WDIM)

// workspace layout (floats)
#define WS_KK2    0
#define WS_S1     (KBINS + 0)
#define WS_S2     (KBINS + 1)
#define WS_COMMIT (KBINS + 2)
#define WS_FIT    (KBINS + 3)

#define AS1 __attribute__((address_space(1)))
#define AS3 __attribute__((address_space(3)))

// ---- async global->LDS copy of 16 bytes (per-lane), ASYNCcnt-tracked -------
#if __has_builtin(__builtin_amdgcn_global_load_async_to_lds_b128)
#define ASYNC_CP16(gsrc, ldst)                                                 \
    __builtin_amdgcn_global_load_async_to_lds_b128(                            \
        (AS1 v4i*)(uintptr_t)(gsrc), (AS3 v4i*)(ldst), 0, 0)
#else
#define ASYNC_CP16(gsrc, ldst)                                                 \
    asm volatile("global_load_async_to_lds_b128 %0, %1, off"                   \
                 :: "v"((AS3 v4i*)(ldst)), "v"((gsrc)) : "memory")
#endif

#if __has_builtin(__builtin_amdgcn_s_wait_asynccnt)
#define WAIT_ASYNC(n) __builtin_amdgcn_s_wait_asynccnt(n)
#else
#define WAIT_ASYNC(n) asm volatile("s_wait_asynccnt %0" :: "n"(n) : "memory")
#endif

// ---------------------------------------------------------------------------
// Kernel 0: precompute ||k_j||^2 and zero scalar accumulators (each launch).
// ---------------------------------------------------------------------------
__global__ void vq_init(const float* __restrict__ Kc, float* __restrict__ ws) {
    int j = blockIdx.x * blockDim.x + threadIdx.x;
    if (j < KBINS) {
        const float* kr = Kc + j * WDIM;
        float s = 0.f;
        #pragma unroll
        for (int w = 0; w < WDIM; ++w) { float v = kr[w]; s += v * v; }
        ws[WS_KK2 + j] = s;
    }
    if (blockIdx.x == 0 && threadIdx.x < 4) ws[KBINS + threadIdx.x] = 0.f;
}

// ---------------------------------------------------------------------------
// Kernel 1: main VQ kernel. 256 threads = 8 waves; each wave: 16 xf rows.
// Double-buffered async codebook staging overlapped with f32 WMMA.
// ---------------------------------------------------------------------------
__global__ __launch_bounds__(256) void vq_main(const float* __restrict__ X,
                                               const float* __restrict__ Kc,
                                               float* __restrict__ out,
                                               float* __restrict__ ws) {
    __shared__ float sB[2][16 * 68];  // double-buffered tile, padded stride
    __shared__ int   sIdx[8][16];     // per-wave argmin rows

    const int tid  = threadIdx.x;
    const int wave = tid >> 5;
    const int lane = tid & 31;
    const int half = lane >> 4;
    const int m    = lane & 15;

    const int r0 = blockIdx.x * ROWS_PER_BLOCK + wave * 16;
    const int n  = r0 >> 12;                 // tiles never cross n boundary
    const int t0 = r0 & 4095;
    const float* Xn = X + n * (WDIM * TDIM);

    // per-thread async staging addresses: 16 B each, 256 threads = 4 KB tile
    const int srow = tid >> 4;               // codebook row within tile
    const int scol = (tid & 15) * 4;         // 16B-aligned column chunk
    const float* gK = Kc + srow * WDIM + scol;
    float* l0 = &sB[0][srow * 68 + scol];
    float* l1 = &sB[1][srow * 68 + scol];

    // --- Preload A fragments (16x4 f32 layout: lanes 0-15 K%4 in {0,1}, 16-31 in {2,3})
    const int koff = half * 2;
    v2f afrag[16];
    float px = 0.f, ps = 0.f;
    #pragma unroll
    for (int s = 0; s < 16; ++s) {
        float a0 = Xn[(4 * s + koff) * TDIM + t0 + m];
        float a1 = Xn[(4 * s + koff + 1) * TDIM + t0 + m];
        afrag[s].x = a0; afrag[s].y = a1;
        px += a0 * a0 + a1 * a1;
        ps += a0 + a1;
    }
    const float xx = px + __shfl_xor(px, 16);

    float minv[8];
    int   mini[8];
    #pragma unroll
    for (int v = 0; v < 8; ++v) { minv[v] = 3.4e38f; mini[v] = 0; }

    // prologue: stage tile 0 into buffer 0
    ASYNC_CP16(gK, l0);

    for (int jt = 0; jt < NTILES; ++jt) {
        const int j0 = jt * 16;
        const int cur = jt & 1;
        // issue next tile into the other buffer (last read 2 barriers ago)
        if (jt + 1 < NTILES) {
            ASYNC_CP16(gK + (jt + 1) * (16 * WDIM), cur ? l0 : l1);
            WAIT_ASYNC(1);                  // in-order: current tile resident
        } else {
            WAIT_ASYNC(0);
        }
        __syncthreads();                     // all waves' tiles visible

        const float* B = sB[cur];
        const float kkn = ws[WS_KK2 + j0 + m];   // ||k_j||^2, L2-hot

        v8f acc = {};
        #pragma unroll
        for (int s = 0; s < 16; ++s) {
            v2f b;
            b.x = B[m * 68 + 4 * s + koff];
            b.y = B[m * 68 + 4 * s + koff + 1];
            acc = __builtin_amdgcn_wmma_f32_16x16x4_f32(
                false, afrag[s], false, b, (short)0, acc, false, false);
        }
        #pragma unroll
        for (int v = 0; v < 8; ++v) {
            float dd = kkn - 2.0f * acc[v];
            if (dd < minv[v]) { minv[v] = dd; mini[v] = j0 + m; }
        }
        __syncthreads();                     // done reading buf before reuse
    }

    // --- Per-row min+argmin reduction across the 16 lanes of each half
    float fitacc = 0.f;
    #pragma unroll
    for (int v = 0; v < 8; ++v) {
        float val = minv[v]; int idx = mini[v];
        #pragma unroll
        for (int off = 1; off < 16; off <<= 1) {
            float ov = __shfl_xor(val, off);
            int   oi = __shfl_xor(idx, off);
            if (ov < val || (ov == val && oi < idx)) { val = ov; idx = oi; }
        }
        const int M = v + half * 8;
        const float xxM = __shfl(xx, M);
        if (m == v) {
            sIdx[wave][M] = idx;
            out[r0 + M] = (float)idx;        // x_l (as f32)
            fitacc += val + xxM;             // min_distance incl. ||x||^2
        }
    }

    __syncthreads();

    // --- Dequantize gather + commit loss; coalesced stores into (N, W, T)
    const int myIdx = sIdx[wave][m];
    float commitacc = 0.f;
    float* Xd = out + XD_OFF + n * (WDIM * TDIM);
    #pragma unroll 4
    for (int ww = 0; ww < 32; ++ww) {
        int w = 2 * ww + half;
        float kv = Kc[myIdx * WDIM + w];
        int off = w * TDIM + t0 + m;
        float xv = Xn[off];
        Xd[off] = kv;
        float df = kv - xv;
        commitacc += df * df;
    }

    // --- Wave-level scalar reductions, one atomicAdd set per wave
    float s1 = ps, s2 = px, cm = commitacc, ft = fitacc;
    #pragma unroll
    for (int off = 16; off >= 1; off >>= 1) {
        s1 += __shfl_xor(s1, off);
        s2 += __shfl_xor(s2, off);
        cm += __shfl_xor(cm, off);
        ft += __shfl_xor(ft, off);
    }
    if (lane == 0) {
        atomicAdd(&ws[WS_S1], s1);
        atomicAdd(&ws[WS_S2], s2);
        atomicAdd(&ws[WS_COMMIT], cm);
        atomicAdd(&ws[WS_FIT], ft);
    }
}

// ---------------------------------------------------------------------------
// Kernel 2: finalize scalars
// ---------------------------------------------------------------------------
__global__ void vq_finalize(const float* __restrict__ ws, float* __restrict__ out) {
    if (blockIdx.x == 0 && threadIdx.x == 0) {
        const float sz = (float)(NROWS * WDIM);
        float S1 = ws[WS_S1], S2 = ws[WS_S2];
        out[SCAL_OFF + 0] = ws[WS_COMMIT] / sz;          // commit_loss
        out[SCAL_OFF + 1] = ws[WS_FIT] / (float)NROWS;   // fit
        float var = (S2 - S1 * S1 / sz) / sz;
        out[SCAL_OFF + 2] = sqrtf(fmaxf(var, 0.f));      // prenorm
    }
}

extern "C" void kernel_launch(void* const* d_in, const int* in_sizes, int n_in,
                              void* d_out, int out_size, void* d_ws, size_t ws_size,
                              hipStream_t stream) {
    const float* X  = (const float*)d_in[0];   // (8, 64, 4096) f32
    const float* Kc = (const float*)d_in[1];   // (2048, 64) f32
    float* out = (float*)d_out;
    float* ws  = (float*)d_ws;

    vq_init<<<(KBINS + 255) / 256, 256, 0, stream>>>(Kc, ws);
    vq_main<<<NROWS / ROWS_PER_BLOCK, 256, 0, stream>>>(X, Kc, out, ws);
    vq_finalize<<<1, 1, 0, stream>>>(ws, out);
}